// SA_4887672783507
// MI455X (gfx1250) — compile-verified
//
#include <hip/hip_runtime.h>
#include <math.h>

// ---------------------------------------------------------------------------
// CDNA5 (gfx1250) implementation of the sliced-transport attention reference.
//   b=2, n=1024, dim=512, heads=8, dim_head=64, L=8, T=0.1
// Output tuple = (out [2,1024,512], attn [2,8,1024,1024]) concatenated in d_out.
// GEMMs use V_WMMA_F32_16X16X4_F32 (fp32 in/out, matches reference precision).
// kp slab staged into LDS via the Tensor Data Mover (tensor_load_to_lds).
// ---------------------------------------------------------------------------

typedef __attribute__((ext_vector_type(2))) float v2f;
typedef __attribute__((ext_vector_type(8))) float v8f;
typedef __attribute__((ext_vector_type(4))) unsigned int u32x4;
typedef __attribute__((ext_vector_type(8))) int i32x8;
typedef __attribute__((ext_vector_type(4))) int i32x4;

// ---- constants for this fixed problem size --------------------------------
#define BB     2
#define NN     1024
#define DIM    512
#define HEADS  8
#define DHEAD  64
#define LSL    8
#define INNER  (HEADS * DHEAD)      // 512
#define QKVW   (3 * INNER)          // 1536
#define INV_T  10.0f                // 1 / temperature
#define LOG2E  1.4426950408889634f

// ---------------------------------------------------------------------------
// Wave-level 16x16 fp32 WMMA tile:  C(16x16) += A(16xK) * B(Kx16), K%4==0.
// A fragment layout (ISA 7.12.2, 32-bit A 16x4): lanes 0-15 carry K={k,k+1}
// for row M=lane; lanes 16-31 carry K={k+2,k+3} for row M=lane-16.
// B fragment assumed symmetric (rows striped across lanes).
// ---------------------------------------------------------------------------
__device__ __forceinline__ v8f wmma_tile_f32(const float* __restrict__ A, int lda,
                                             const float* __restrict__ B, int ldb,
                                             int K) {
    const int lane = threadIdx.x & 31;
    const bool lo  = lane < 16;
    const int  r   = lane & 15;
    const float* arow = A + (size_t)r * lda + (lo ? 0 : 2);
    const float* bcol = B + r;
    const int brow0 = lo ? 0 : 2;
    v8f c = {};
#pragma unroll 2
    for (int k = 0; k < K; k += 4) {
        // prefetch next A panel into cache (global_prefetch_b8)
        if (k + 64 < K) __builtin_prefetch(arow + k + 64, 0, 1);
        v2f a = *(const v2f*)(arow + k);           // A[m][k+off], A[m][k+off+1]
        v2f b;
        b.x = bcol[(size_t)(k + brow0    ) * ldb]; // B[k+off  ][col]
        b.y = bcol[(size_t)(k + brow0 + 1) * ldb]; // B[k+off+1][col]
        c = __builtin_amdgcn_wmma_f32_16x16x4_f32(false, a, false, b,
                                                  (short)0, c, false, false);
    }
    return c;
}

// C/D layout: VGPR r -> row (tm + r + (lane<16?0:8)), col = tn + (lane&15)
__device__ __forceinline__ void store_tile_f32(float* __restrict__ C, int ldc,
                                               v8f c, const float* __restrict__ bias) {
    const int lane = threadIdx.x & 31;
    const int r0   = (lane < 16) ? 0 : 8;
    const int col  = lane & 15;
    const float bv = bias ? bias[col] : 0.0f;
#pragma unroll
    for (int r = 0; r < 8; ++r)
        C[(size_t)(r0 + r) * ldc + col] = c[r] + bv;
}

// ---------------------------------------------------------------------------
// Plain GEMM: C[M,N] = A[M,K] @ B[K,N] (+bias per column). One wave per tile.
// ---------------------------------------------------------------------------
__global__ __launch_bounds__(32)
void gemm_wmma_f32(const float* __restrict__ A, const float* __restrict__ B,
                   float* __restrict__ C, const float* __restrict__ bias,
                   int K, int lda, int ldb, int ldc) {
    const int tn = blockIdx.x * 16;
    const int tm = blockIdx.y * 16;
    v8f c = wmma_tile_f32(A + (size_t)tm * lda, lda, B + tn, ldb, K);
    store_tile_f32(C + (size_t)tm * ldc + tn, ldc, c, bias ? bias + tn : nullptr);
}

// ---------------------------------------------------------------------------
// qp/kp slice projections. theta row normalized on the fly; scale=1/sqrt(64)
// folded into qp and kp (reference scales q and k before projecting).
//   qp[((bb*H+h)*L+l)*n + i] = scale * sum_d qkv[bb,i, h*64+d      ] * thn[d]
//   kp[...]                  = scale * sum_d qkv[bb,i, 512+h*64+d  ] * thn[d]
// ---------------------------------------------------------------------------
__global__ __launch_bounds__(256)
void qkp_proj(const float* __restrict__ qkv, const float* __restrict__ theta,
              float* __restrict__ qp, float* __restrict__ kp) {
    const int s  = blockIdx.x;            // 0..127 = ((bb*8+h)*8+l)
    const int l  = s & 7;
    const int h  = (s >> 3) & 7;
    const int bb = s >> 6;
    const int i  = blockIdx.y * 256 + threadIdx.x;

    const float* th = theta + ((size_t)h * LSL + l) * DHEAD;
    float nrm = 0.0f;
#pragma unroll
    for (int d = 0; d < DHEAD; ++d) nrm += th[d] * th[d];
    nrm = rsqrtf(nrm);

    const float* qrow = qkv + ((size_t)bb * NN + i) * QKVW + h * DHEAD;
    const float* krow = qrow + INNER;
    float dq = 0.0f, dk = 0.0f;
#pragma unroll 8
    for (int d = 0; d < DHEAD; ++d) {
        const float w = th[d] * nrm;
        dq += qrow[d] * w;
        dk += krow[d] * w;
    }
    const float scale = 0.125f;           // 64^-0.5
    qp[(size_t)s * NN + i] = dq * scale;
    kp[(size_t)s * NN + i] = dk * scale;
}

// ---------------------------------------------------------------------------
// attn[b,h,i,j] = (1/L) * sum_l softmax_j( -(qp_l[i]-kp_l[j])^2 / T )
// One block per (b*h, 64-row slab). kp[L=8][n=1024] (32 KB of the 320 KB WGP
// LDS pool) is DMA'd in by the Tensor Data Mover: wave 0 issues a single
// tensor_load_to_lds (1D tile: data_size=4B, dim0 = tile_dim0 = stride = 8192
// elements), waits on TENSORcnt, and the workgroup barrier releases the rest.
// The TDM write is invisible to LLVM alias analysis, so the LDS pointer is
// escaped through an asm memory clobber to keep the ds_load path alive.
// Each wave32 then owns one row at a time; each lane owns 32 j's kept in
// registers across the two passes (min-shift then exp/normalize).
// ---------------------------------------------------------------------------
__global__ __launch_bounds__(128)
void attn_rows(const float* __restrict__ qp, const float* __restrict__ kp,
               float* __restrict__ attn) {
    __shared__ float kps[LSL][NN];        // 32 KB; sole LDS object -> offset 0
    const int bh = blockIdx.x;            // 0..15

    if (threadIdx.x < 32) {               // wave 0 issues the TDM descriptor
        const float* src = kp + (size_t)bh * LSL * NN;
        const unsigned long long ga = (unsigned long long)(uintptr_t)src;
        // D# group 0: count=1 (valid user descriptor), lds_addr=0,
        //             global_addr[56:0], type=2 ("image") in bits 127:126
        u32x4 g0;
        g0[0] = 1u;
        g0[1] = 0u;
        g0[2] = (unsigned int)(ga & 0xFFFFFFFFu);
        g0[3] = (unsigned int)((ga >> 32) & 0x1FFFFFFu) | (2u << 30);
        // D# group 1: workgroup_mask=0 (not in cluster), data_size=2 (4 bytes),
        //             tensor_dim0 = 8192 (bits 79:48), tile_dim0 = 8192
        //             (bits 127:112), tensor_dim0_stride = 8192 (bits 207:160)
        i32x8 g1;
        g1[0] = (int)(2u << 16);
        g1[1] = (int)(8192u << 16);       // tensor_dim0 low 16 bits
        g1[2] = 0;                        // tensor_dim0 hi / tensor_dim1 = 0
        g1[3] = (int)(8192u << 16);       // tile_dim0
        g1[4] = 0;                        // tile_dim1/2 unused
        g1[5] = 8192;                     // tensor_dim0_stride low 32 bits
        g1[6] = 0;
        g1[7] = 0;
        i32x4 z4 = {0, 0, 0, 0};          // groups 2/3 unused (<=2D tensor)
        i32x8 z8 = {0, 0, 0, 0, 0, 0, 0, 0};
        __builtin_amdgcn_tensor_load_to_lds(g0, g1, z4, z4, z8, 0);
        __builtin_amdgcn_s_wait_tensorcnt(0);
    }
    // Make the DMA's LDS write visible to the compiler: escape the pointer and
    // clobber memory so kps loads cannot be folded to undef / eliminated.
    {
        float* kps_escape = &kps[0][0];
        asm volatile("" : "+v"(kps_escape) : : "memory");
    }
    __syncthreads();

    const int wave = threadIdx.x >> 5;
    const int lane = threadIdx.x & 31;

    for (int rr = 0; rr < 16; ++rr) {
        const int i = blockIdx.y * 64 + rr * 4 + wave;
        float acc[32];
#pragma unroll
        for (int t = 0; t < 32; ++t) acc[t] = 0.0f;
        float e[32];

#pragma unroll 1
        for (int l = 0; l < LSL; ++l) {
            const float qpi = qp[((size_t)bh * LSL + l) * NN + i];
            // pass 1: min over j of (qp_i - kp_j)^2  (== max of the logits)
            float dmin = 3.4e38f;
#pragma unroll
            for (int t = 0; t < 32; ++t) {
                float d = qpi - kps[l][lane + 32 * t];
                d *= d;
                dmin = fminf(dmin, d);
            }
#pragma unroll
            for (int m = 16; m >= 1; m >>= 1)
                dmin = fminf(dmin, __shfl_xor(dmin, m, 32));
            // pass 2: exp and row sum
            const float cexp = -INV_T * LOG2E;     // exp(-(d2-dmin)/T) via exp2
            float ssum = 0.0f;
#pragma unroll
            for (int t = 0; t < 32; ++t) {
                float d = qpi - kps[l][lane + 32 * t];
                d *= d;
                const float ev = exp2f((d - dmin) * cexp);
                e[t] = ev;
                ssum += ev;
            }
#pragma unroll
            for (int m = 16; m >= 1; m >>= 1)
                ssum += __shfl_xor(ssum, m, 32);
            const float inv = 1.0f / ssum;
#pragma unroll
            for (int t = 0; t < 32; ++t) acc[t] += e[t] * inv;
        }

        float* arow = attn + ((size_t)bh * NN + i) * NN;
#pragma unroll
        for (int t = 0; t < 32; ++t)
            arow[lane + 32 * t] = acc[t] * (1.0f / LSL);
    }
}

// ---------------------------------------------------------------------------
// out_h = attn @ v, batched over (bb,h); result scattered directly into
// [b, n, h*64+d] layout so the final projection is a plain GEMM.
//   A = attn[(bb*8+h)] : [1024,1024]      lda = 1024
//   B = qkv[bb, :, 1024 + h*64 + d]       ldb = 1536
//   C = outh[bb, i, h*64 + d]             ldc = 512
// ---------------------------------------------------------------------------
__global__ __launch_bounds__(32)
void attnv_wmma(const float* __restrict__ attn, const float* __restrict__ qkv,
                float* __restrict__ outh) {
    const int t  = blockIdx.z;            // 0..15
    const int bb = t >> 3;
    const int h  = t & 7;
    const int tn = blockIdx.x * 16;       // 0..48  (dhead)
    const int tm = blockIdx.y * 16;       // 0..1008 (rows)
    const float* A = attn + (size_t)t * NN * NN + (size_t)tm * NN;
    const float* B = qkv + (size_t)bb * NN * QKVW + 2 * INNER + h * DHEAD + tn;
    float*       C = outh + ((size_t)bb * NN + tm) * INNER + h * DHEAD + tn;
    v8f c = wmma_tile_f32(A, NN, B, QKVW, NN);
    store_tile_f32(C, INNER, c, nullptr);
}

// ---------------------------------------------------------------------------
// host-side launch
// ---------------------------------------------------------------------------
extern "C" void kernel_launch(void* const* d_in, const int* in_sizes, int n_in,
                              void* d_out, int out_size, void* d_ws, size_t ws_size,
                              hipStream_t stream) {
    const float* x     = (const float*)d_in[0];   // [2,1024,512]
    const float* W_qkv = (const float*)d_in[1];   // [512,1536]
    const float* theta = (const float*)d_in[2];   // [8,8,64]
    const float* W_out = (const float*)d_in[3];   // [512,512]
    const float* b_out = (const float*)d_in[4];   // [512]

    float* out  = (float*)d_out;                        // [2,1024,512]
    float* attn = out + (size_t)BB * NN * DIM;          // [2,8,1024,1024]

    // workspace carve-up (~17.8 MB total)
    char*  ws     = (char*)d_ws;
    float* qkvbuf = (float*)ws;                                  // 2*1024*1536
    float* qp     = qkvbuf + (size_t)BB * NN * QKVW;             // 2*8*8*1024
    float* kp     = qp + (size_t)BB * HEADS * LSL * NN;
    float* outh   = kp + (size_t)BB * HEADS * LSL * NN;          // 2*1024*512

    // 1) qkv = x @ W_qkv   (M=2048, N=1536, K=512)
    gemm_wmma_f32<<<dim3(QKVW / 16, (BB * NN) / 16), 32, 0, stream>>>(
        x, W_qkv, qkvbuf, nullptr, DIM, DIM, QKVW, QKVW);

    // 2) slice projections qp/kp
    qkp_proj<<<dim3(BB * HEADS * LSL, NN / 256), 256, 0, stream>>>(
        qkvbuf, theta, qp, kp);

    // 3) transport-plan softmax -> attn (written directly into d_out tail)
    attn_rows<<<dim3(BB * HEADS, NN / 64), 128, 0, stream>>>(qp, kp, attn);

    // 4) out_h = attn @ v  (batched, scattered into [b,n,inner] layout)
    attnv_wmma<<<dim3(DHEAD / 16, NN / 16, BB * HEADS), 32, 0, stream>>>(
        attn, qkvbuf, outh);

    // 5) out = out_h @ W_out + b_out   (M=2048, N=512, K=512)
    gemm_wmma_f32<<<dim3(DIM / 16, (BB * NN) / 16), 32, 0, stream>>>(
        outh, W_out, out, b_out, INNER, INNER, DIM, DIM);
}